// ConcatenateSphericalSignals_46428596469936
// MI455X (gfx1250) — compile-verified
//
#include <hip/hip_runtime.h>

// ConcatenateSphericalSignals on MI455X (gfx1250).
//
// out[n,d,b,a] = sum_c M[d,c] * concat(s1,s2)[n,c,b,a], M a one-hot permutation
// => pure channel gather, memory-bound (~384 MiB moved -> ~17 us at 23.3 TB/s).
// Implemented with the CDNA5 async global<->LDS path (ASYNCcnt-tracked
// GLOBAL_LOAD_ASYNC_TO_LDS_B128 / GLOBAL_STORE_ASYNC_FROM_LDS_B128).

#define F1C 288
#define F2C 480
#define CC  768    // F1C + F2C
#define BA  4096   // 64 * 64
#define NT  256    // threads per block (8 wave32)
#define ROUNDS 4   // 256 lanes * 16 B * 4 rounds = 16 KB slice

__global__ __launch_bounds__(NT) void permute_channels_async(
    const float* __restrict__ s1,
    const float* __restrict__ s2,
    const float* __restrict__ M,
    float* __restrict__ out)
{
    __shared__ float lds_buf[BA];   // 16 KB staging buffer
    __shared__ int   s_src;

    const int d = blockIdx.x;       // output channel 0..767
    const int n = blockIdx.y;       // batch 0..15
    const int t = threadIdx.x;

    // Row d of the one-hot matrix has exactly one nonzero: find its column.
    const float* Mrow = M + (size_t)d * CC;
    for (int c = t; c < CC; c += NT) {
        if (Mrow[c] != 0.0f) s_src = c;   // exactly one lane writes
    }
    __syncthreads();
    const int c = s_src;

    const float* src_base = (c < F1C)
        ? (s1 + ((size_t)n * F1C + c) * BA)
        : (s2 + ((size_t)n * F2C + (c - F1C)) * BA);
    float* dst_base = out + ((size_t)n * CC + d) * BA;

    // LDS byte address = low 32 bits of the generic (flat) pointer.
    const uint32_t lds0 = (uint32_t)(size_t)(void*)lds_buf;

    // Stage 1: async global -> LDS, 16 B per lane per round.
    // Each lane's LDS region is private, so only a per-wave asynccnt wait
    // is needed before the stores (no workgroup barrier).
#pragma unroll
    for (int r = 0; r < ROUNDS; ++r) {
        const uint32_t off = (uint32_t)(t * 16 + r * (NT * 16));
        const uint32_t la  = lds0 + off;
        asm volatile("global_load_async_to_lds_b128 %0, %1, %2 offset:0"
                     :: "v"(la), "v"(off), "s"(src_base)
                     : "memory");
    }

    asm volatile("s_wait_asynccnt 0x0" ::: "memory");

    // Stage 2: async LDS -> global. S_ENDPGM implicitly waits all counters.
#pragma unroll
    for (int r = 0; r < ROUNDS; ++r) {
        const uint32_t off = (uint32_t)(t * 16 + r * (NT * 16));
        const uint32_t la  = lds0 + off;
        asm volatile("global_store_async_from_lds_b128 %0, %1, %2 offset:0"
                     :: "v"(off), "v"(la), "s"(dst_base)
                     : "memory");
    }
}

extern "C" void kernel_launch(void* const* d_in, const int* in_sizes, int n_in,
                              void* d_out, int out_size, void* d_ws, size_t ws_size,
                              hipStream_t stream)
{
    const float* signal1 = (const float*)d_in[0];   // [16, 288, 64, 64]
    const float* signal2 = (const float*)d_in[1];   // [16, 480, 64, 64]
    const float* mixing  = (const float*)d_in[2];   // [768, 768] one-hot
    float* out = (float*)d_out;                     // [16, 768, 64, 64]

    (void)in_sizes; (void)n_in; (void)out_size; (void)d_ws; (void)ws_size;

    dim3 grid(CC, 16, 1);   // one block per (d, n) slice of 16 KB
    dim3 block(NT, 1, 1);
    permute_channels_async<<<grid, block, 0, stream>>>(signal1, signal2, mixing, out);
}